// OTSolver_60679297958190
// MI455X (gfx1250) — compile-verified
//
#include <hip/hip_runtime.h>
#include <math.h>

// Sinkhorn-Knopp OT solver for MI455X (gfx1250, wave32).
// K (64 MB fp32) is staged in the d_out P-slot and is L2-resident (192 MB L2),
// so all 200 GEMV sweeps stream from L2. GEMVs run on V_WMMA_F32_16X16X4_F32
// with the vector operand replicated across the 16-wide free dimension:
// full fp32 precision, FMA work on the matrix pipe, memory-roof bound.

typedef float v2f __attribute__((ext_vector_type(2)));
typedef float v8f __attribute__((ext_vector_type(8)));

#define SK_N     4096
#define SK_NN    (4096 * 4096)
#define SK_IEPS  10.0f      // 1/EPSILON
#define SK_TOL   1e-9f
#define SK_ITERS 100

// ---------------------------------------------------------------- init K
__global__ void __launch_bounds__(256) sk_init_K(const float* __restrict__ C,
                                                 float* __restrict__ K) {
  size_t i4 = ((size_t)blockIdx.x * 256 + threadIdx.x) * 4;
  float4 c4 = *(const float4*)(C + i4);
  float4 k4;
  k4.x = expf(-SK_IEPS * c4.x);
  k4.y = expf(-SK_IEPS * c4.y);
  k4.z = expf(-SK_IEPS * c4.z);
  k4.w = expf(-SK_IEPS * c4.w);
  *(float4*)(K + i4) = k4;
}

__global__ void sk_ones(float* __restrict__ v) {
  v[blockIdx.x * 256 + threadIdx.x] = 1.0f;
}

// ------------------------------------------------- acc[i] = sum_j K[i][j]*v[j]
// Block = 8 waves, owns 16 rows x all 4096 cols. Wave w covers cols
// [512w, 512w+512). A = 16x4 tile of K (ISA striped layout), B = v replicated
// across the 16 free columns, C accumulates in fp32.
__global__ void __launch_bounds__(256) sk_matvec_Kv(const float* __restrict__ K,
                                                    const float* __restrict__ v,
                                                    float* __restrict__ acc_out) {
  __shared__ float part[8][16];
  const int wave = threadIdx.x >> 5;
  const int lane = threadIdx.x & 31;
  const int hf   = lane >> 4;            // 0: K=0,1  1: K=2,3  (ISA A layout)
  const int r    = lane & 15;            // matrix row within tile
  const int rowBase = blockIdx.x << 4;
  const float* Krow = K + (size_t)(rowBase + r) * SK_N + 2 * hf;
  const float* vp   = v + 2 * hf;

  v8f acc = {0.f, 0.f, 0.f, 0.f, 0.f, 0.f, 0.f, 0.f};
  const int c0 = wave << 9;
#pragma unroll 4
  for (int c = c0; c < c0 + 512; c += 4) {
    v2f a = *(const v2f*)(Krow + c);     // A[r][k], k per ISA lane layout
    v2f b = *(const v2f*)(vp + c);       // B[k][n] = v[c+k]  (replicated in n)
    acc = __builtin_amdgcn_wmma_f32_16x16x4_f32(false, a, false, b,
                                                (short)0, acc, false, false);
  }
  // D columns are identical; lane 0 holds rows 0-7, lane 16 holds rows 8-15.
  if (r == 0) {
    float* dst = &part[wave][hf * 8];
    dst[0] = acc[0]; dst[1] = acc[1]; dst[2] = acc[2]; dst[3] = acc[3];
    dst[4] = acc[4]; dst[5] = acc[5]; dst[6] = acc[6]; dst[7] = acc[7];
  }
  __syncthreads();
  if (threadIdx.x < 16) {
    float s = 0.f;
#pragma unroll
    for (int w = 0; w < 8; ++w) s += part[w][threadIdx.x];
    acc_out[rowBase + threadIdx.x] = s;
  }
}

// ------------------------------------------------- acc[j] = sum_i K[i][j]*u[i]
// Block owns 16 cols x all 4096 rows. A = u replicated over the 16 free rows,
// B = 4x16 tile of K (64B contiguous per tile row -> coalesced per half-wave).
__global__ void __launch_bounds__(256) sk_matvec_KT(const float* __restrict__ K,
                                                    const float* __restrict__ u,
                                                    float* __restrict__ acc_out) {
  __shared__ float part[8][16];
  const int wave = threadIdx.x >> 5;
  const int lane = threadIdx.x & 31;
  const int hf   = lane >> 4;
  const int r    = lane & 15;
  const int colBase = blockIdx.x << 4;
  const float* Kc = K + colBase + r;

  v8f acc = {0.f, 0.f, 0.f, 0.f, 0.f, 0.f, 0.f, 0.f};
  const int r0 = wave << 9;
#pragma unroll 2
  for (int row = r0; row < r0 + 512; row += 4) {
    v2f a = *(const v2f*)(u + row + 2 * hf);         // A[m][k]=u[row+k], any m
    size_t base = (size_t)(row + 2 * hf) * SK_N;
    v2f b;
    b[0] = Kc[base];                                 // B[k][n]   = K[row+k][col+n]
    b[1] = Kc[base + SK_N];                          // B[k+1][n]
    acc = __builtin_amdgcn_wmma_f32_16x16x4_f32(false, a, false, b,
                                                (short)0, acc, false, false);
  }
  // All D rows identical; VGPR0 @ lanes 0-15 is row 0, col = lane.
  if (lane < 16) part[wave][lane] = acc[0];
  __syncthreads();
  if (threadIdx.x < 16) {
    float s = 0.f;
#pragma unroll
    for (int w = 0; w < 8; ++w) s += part[w][threadIdx.x];
    acc_out[colBase + threadIdx.x] = s;
  }
}

// ---------------------------------------------- x[i] = target[i]/(acc[i]+tol)
__global__ void sk_scale(const float* __restrict__ target,
                         const float* __restrict__ acc,
                         float* __restrict__ out) {
  int i = blockIdx.x * 256 + threadIdx.x;
  out[i] = target[i] / (acc[i] + SK_TOL);
}

// ------------------- P[i][j] = u_i*K_ij*v_j (in place), row loss partials
__global__ void __launch_bounds__(256) sk_final(const float* __restrict__ mu,
                                                const float* __restrict__ nu,
                                                const float* __restrict__ u,
                                                const float* __restrict__ v,
                                                float* __restrict__ P,
                                                float* __restrict__ lpart) {
  __shared__ float red[256];
  const int i = blockIdx.x;
  const float ui = u[i];
  const float mi = mu[i];
  float* Prow = P + (size_t)i * SK_N;
  float w = 0.f;
  for (int j = threadIdx.x * 4; j < SK_N; j += 256 * 4) {
    float4 k4 = *(const float4*)(Prow + j);
    float4 v4 = *(const float4*)(v + j);
    float4 n4 = *(const float4*)(nu + j);
    float4 p;
    p.x = ui * k4.x * v4.x;
    p.y = ui * k4.y * v4.y;
    p.z = ui * k4.z * v4.z;
    p.w = ui * k4.w * v4.w;
    *(float4*)(Prow + j) = p;
    w += p.x * fabsf(mi - n4.x) + p.y * fabsf(mi - n4.y) +
         p.z * fabsf(mi - n4.z) + p.w * fabsf(mi - n4.w);
  }
  red[threadIdx.x] = w;
  __syncthreads();
  for (int s = 128; s > 0; s >>= 1) {
    if (threadIdx.x < s) red[threadIdx.x] += red[threadIdx.x + s];
    __syncthreads();
  }
  if (threadIdx.x == 0) lpart[i] = red[0];
}

__global__ void __launch_bounds__(256) sk_loss_reduce(const float* __restrict__ lpart,
                                                      float* __restrict__ loss) {
  __shared__ float red[256];
  float s = 0.f;
  for (int i = threadIdx.x; i < SK_N; i += 256) s += lpart[i];
  red[threadIdx.x] = s;
  __syncthreads();
  for (int k = 128; k > 0; k >>= 1) {
    if (threadIdx.x < k) red[threadIdx.x] += red[threadIdx.x + k];
    __syncthreads();
  }
  if (threadIdx.x == 0) *loss = red[0];
}

// ----------------------------------------------------------------- launcher
extern "C" void kernel_launch(void* const* d_in, const int* in_sizes, int n_in,
                              void* d_out, int out_size, void* d_ws, size_t ws_size,
                              hipStream_t stream) {
  const float* mu = (const float*)d_in[0];
  const float* nu = (const float*)d_in[1];
  const float* C  = (const float*)d_in[2];
  float* K = (float*)d_out;              // K staged in the P output slot
  float* ws = (float*)d_ws;
  float* u     = ws;                     // N
  float* v     = ws + SK_N;              // N
  float* acc   = ws + 2 * SK_N;          // N (shared between Kv and KTu)
  float* lpart = ws + 3 * SK_N;          // N

  sk_init_K<<<SK_NN / (256 * 4), 256, 0, stream>>>(C, K);
  sk_ones<<<SK_N / 256, 256, 0, stream>>>(v);

  for (int t = 0; t < SK_ITERS; ++t) {
    sk_matvec_Kv<<<SK_N / 16, 256, 0, stream>>>(K, v, acc);
    sk_scale<<<SK_N / 256, 256, 0, stream>>>(mu, acc, u);
    sk_matvec_KT<<<SK_N / 16, 256, 0, stream>>>(K, u, acc);
    sk_scale<<<SK_N / 256, 256, 0, stream>>>(nu, acc, v);
  }

  sk_final<<<SK_N, 256, 0, stream>>>(mu, nu, u, v, K, lpart);
  sk_loss_reduce<<<1, 256, 0, stream>>>(lpart, K + SK_NN);
}